// HGTCavAttention_5669356834454
// MI455X (gfx1250) — compile-verified
//
#include <hip/hip_runtime.h>

// ---------------------------------------------------------------------------
// HGT cavity attention for MI455X (gfx1250, wave32).
// All contractions use V_WMMA_F32_16X16X4_F32 (exact fp32, matches reference).
// A-tiles of the two big projections are staged LDS-ward by the Tensor Data
// Mover (tensor_load_to_lds + s_wait_tensorcnt), using TDM padding to produce
// the bank-conflict-free LDS stride directly.
//
// f32 WMMA fragment conventions used (per CDNA5 ISA 7.12.2):
//   A (16x4):  lane = m + 16*hi ; vgpr r holds A[m][2*hi + r]   (float2/lane)
//   B (4x16):  lane = n + 16*hi ; vgpr r holds B[2*hi + r][n]
//   C/D(16x16):lane = n + 16*hi ; vgpr r holds C[r + 8*hi][n]
// B tiles are stored k-pair-major in LDS so each B fragment is ONE ds_load_b64.
// ---------------------------------------------------------------------------

typedef float v2f __attribute__((ext_vector_type(2)));
typedef float v8f __attribute__((ext_vector_type(8)));
typedef unsigned int u32x4 __attribute__((ext_vector_type(4)));
typedef int i32x4 __attribute__((ext_vector_type(4)));
typedef int i32x8 __attribute__((ext_vector_type(8)));

#define WMMA_F32(a, b, c) \
  __builtin_amdgcn_wmma_f32_16x16x4_f32(false, (a), false, (b), (short)0, (c), false, false)

static constexpr int kB     = 2;
static constexpr int kL     = 5;
static constexpr int kHW    = 64 * 64;   // 4096 pixels
static constexpr int kDim   = 256;
static constexpr int kInner = 256;
static constexpr int kHeads = 8;
static constexpr int kDH    = 32;
static constexpr int kNT    = 2;
static constexpr int kNRel  = 4;

__device__ __forceinline__ v8f zero8() {
  v8f z = {0.f, 0.f, 0.f, 0.f, 0.f, 0.f, 0.f, 0.f};
  return z;
}

// Low 32 bits of a generic LDS address == byte offset within the wave's LDS.
__device__ __forceinline__ unsigned int lds_offset(const void* p) {
  return (unsigned int)(uintptr_t)p;
}

// ---------------------------------------------------------------------------
// TDM: async-copy a 2D tile [tileH rows x 32 f32] (row stride rowStrideElems
// in global, f32 elements) into LDS with row stride 36 dwords (pad 4 dwords
// every 32: pad_interval=4 (32 dw), pad_amount=3 (4 dw)).  ISA 08 §8.3/8.4.
// ---------------------------------------------------------------------------
__device__ __forceinline__ void tdm_load_a_tile(unsigned int ldsByteOff,
                                                const float* gsrc,
                                                unsigned int rowStrideElems,
                                                unsigned int tileH) {
  const unsigned long long ga = (unsigned long long)(uintptr_t)gsrc;
  u32x4 g0;
  g0[0] = 1u;                                       // count=1, user descriptor
  g0[1] = ldsByteOff;                               // lds_addr (bytes)
  g0[2] = (unsigned int)(ga & 0xFFFFFFFFull);       // global_addr[31:0]
  g0[3] = ((unsigned int)(ga >> 32) & 0x01FFFFFFu)  // global_addr[56:32]
          | 0x80000000u;                            // type = 2 ("image")
  i32x8 g1;
  g1[0] = (int)((2u << 16)      // data_size = 4B
                | (1u << 20)    // pad_enable
                | (4u << 22)    // pad_interval: 32 dwords
                | (3u << 25));  // pad_amount:   4 dwords
  g1[1] = (int)((rowStrideElems & 0xFFFFu) << 16);  // tensor_dim0[15:0]
  g1[2] = (int)(((rowStrideElems >> 16) & 0xFFFFu)  // tensor_dim0[31:16]
                | ((tileH & 0xFFFFu) << 16));       // tensor_dim1[15:0]
  g1[3] = (int)(32u << 16);                         // tile_dim0 = 32 elems
  g1[4] = (int)(tileH & 0xFFFFu);                   // tile_dim1 (tile_dim2 = 0)
  g1[5] = (int)rowStrideElems;                      // tensor_dim0_stride[31:0]
  g1[6] = 0;                                        // stride hi / dim1_stride lo
  g1[7] = 0;
  const i32x4 z4 = {0, 0, 0, 0};
#if defined(__clang_major__) && (__clang_major__ >= 23)
  const i32x8 z8 = {0, 0, 0, 0, 0, 0, 0, 0};
  __builtin_amdgcn_tensor_load_to_lds(g0, g1, z4, z4, z8, 0);
#else
  __builtin_amdgcn_tensor_load_to_lds(g0, g1, z4, z4, 0);
#endif
}

// ---------------------------------------------------------------------------
// Kernel 1: per-type Q/K/V projection.
//   For fixed (b,l): A = x[b,l,:,:,:] contiguous [4096 x 256], W = W*[type(b,l)].
//   Output layout q/k/v: [B][HW][L][INNER]  (attention-friendly).
//   Block: 256 thr (8 waves) -> 64x256 tile; wave = (rowTile, 8 colTiles).
// ---------------------------------------------------------------------------
__global__ __launch_bounds__(256) void qkv_proj_kernel(
    const float* __restrict__ x, const float* __restrict__ pe,
    const float* __restrict__ Wq, const float* __restrict__ bq, float* __restrict__ oq,
    const float* __restrict__ Wk, const float* __restrict__ bk, float* __restrict__ ok,
    const float* __restrict__ Wv, const float* __restrict__ bv, float* __restrict__ ov)
{
  constexpr int AS = 36;   // A-tile LDS row stride (TDM pad; conflict-free, 16B ok)
  constexpr int BS = 34;   // B-tile per-column stride, k-pair-major
  __shared__ __align__(16) float Alds[64 * AS];
  __shared__ __align__(16) float Blds[256 * BS];

  const int bl   = blockIdx.y;
  const int bb   = bl / kL, ll = bl % kL;
  const int row0 = blockIdx.x * 64;
  const int typ  = (int)pe[(bb * kL + ll) * 3 + 2];

  const float* W;  const float* bias;  float* outp;
  if (blockIdx.z == 0)      { W = Wq; bias = bq; outp = oq; }
  else if (blockIdx.z == 1) { W = Wk; bias = bk; outp = ok; }
  else                      { W = Wv; bias = bv; outp = ov; }
  W    += (size_t)typ * kDim * kInner;
  bias += typ * kInner;

  const int tid  = threadIdx.x;
  const int wave = tid >> 5, lane = tid & 31;
  const int rt     = wave & 3;             // row tile 0..3
  const int ntBase = (wave >> 2) * 8;      // first of 8 col tiles
  const int mm = lane & 15, hi = lane >> 4, kb2 = hi * 2;

  const float* Abase = x + ((size_t)(bb * kL + ll) * kHW + row0) * kDim;
  const unsigned int aLds = lds_offset(&Alds[0]);

  v8f acc[8];
  #pragma unroll
  for (int t = 0; t < 8; ++t) acc[t] = zero8();

  for (int kb = 0; kb < kDim; kb += 32) {
    __syncthreads();
    // TDM: async A tile (64 x 32 f32) -> Alds with padded stride 36.
    if (wave == 0) tdm_load_a_tile(aLds, Abase + kb, kDim, 64);
    // Meanwhile all waves stage W chunk transposed, k-pair-major: WT[n][k].
    for (int u = tid; u < 16 * 64; u += 256) {       // (kpair, 4-col group)
      const int kp = u >> 6, c4 = (u & 63) << 2, k2 = kp * 2;
      const float4 r0 = *(const float4*)(W + (size_t)(kb + k2) * kInner + c4);
      const float4 r1 = *(const float4*)(W + (size_t)(kb + k2 + 1) * kInner + c4);
      v2f p0 = {r0.x, r1.x}, p1 = {r0.y, r1.y}, p2 = {r0.z, r1.z}, p3 = {r0.w, r1.w};
      *(v2f*)(&Blds[(c4 + 0) * BS + k2]) = p0;
      *(v2f*)(&Blds[(c4 + 1) * BS + k2]) = p1;
      *(v2f*)(&Blds[(c4 + 2) * BS + k2]) = p2;
      *(v2f*)(&Blds[(c4 + 3) * BS + k2]) = p3;
    }
    if (wave == 0) __builtin_amdgcn_s_wait_tensorcnt(0);
    __syncthreads();

    #pragma unroll
    for (int kk = 0; kk < 32; kk += 4) {
      const v2f a = *(const v2f*)(&Alds[(rt * 16 + mm) * AS + kk + kb2]);
      #pragma unroll
      for (int t = 0; t < 8; ++t) {
        const int ncol = (ntBase + t) * 16 + mm;
        const v2f bf = *(const v2f*)(&Blds[ncol * BS + kk + kb2]);  // one b64
        acc[t] = WMMA_F32(a, bf, acc[t]);
      }
    }
  }

  // Epilogue: +bias, scatter to [B][HW][L][INNER].
  #pragma unroll
  for (int t = 0; t < 8; ++t) {
    const int col = (ntBase + t) * 16 + mm;
    const float bia = bias[col];
    #pragma unroll
    for (int r = 0; r < 8; ++r) {
      const int row = row0 + rt * 16 + r + 8 * hi;
      outp[((size_t)(bb * kHW + row) * kL + ll) * kInner + col] = acc[t][r] + bia;
    }
  }
}

// ---------------------------------------------------------------------------
// Kernel 2: attention core. Block = (b, head, 16-pixel tile), 8 waves.
//   Phase 1: t = k_j * rel_att[e(i,j)]^T via WMMA; att = <q_i, t> via shfl-reduce.
//   Phase 2: mask + softmax over j.
//   Phase 3: msg = v_j * rel_msg[e(i,j)] via WMMA, accumulate att-weighted.
//   Writes merged-heads buffer hm[B][HW][L][INNER] (aliases q_ws: each block
//   overwrites exactly the q region it already consumed; disjoint across blocks).
// ---------------------------------------------------------------------------
__global__ __launch_bounds__(256) void attn_kernel(
    const float* __restrict__ q, const float* __restrict__ k, const float* __restrict__ v,
    const float* __restrict__ mask, const float* __restrict__ pe,
    const float* __restrict__ rel_att, const float* __restrict__ rel_msg,
    float* __restrict__ hm)
{
  constexpr int TS = 162;  // pixel stride in k/v tiles (pad 2: conflict-free float2 frags)
  constexpr int WS = 34;   // per-column stride in rel-weight LDS (k-pair-major)
  __shared__ __align__(16) float WatL[kNRel][kDH * WS]; // rel_att direct [p][q]: q-pairs contig
  __shared__ __align__(16) float WmT [kNRel][kDH * WS]; // rel_msg transposed [c][p]
  __shared__ __align__(16) float KT[16 * TS];
  __shared__ __align__(16) float VT[16 * TS];
  __shared__ float attS[16][kL][kL];
  __shared__ int   typesS[kL];

  const int bb   = blockIdx.z;
  const int mh   = blockIdx.y;        // head
  const int pix0 = blockIdx.x * 16;
  const int tid  = threadIdx.x, wave = tid >> 5, lane = tid & 31;
  const int mm = lane & 15, hi = lane >> 4, kb2 = hi * 2;

  if (tid < kL) typesS[tid] = (int)pe[(bb * kL + tid) * 3 + 2];

  // rel weights for this head (4 edge types x 32x32 each).
  for (int u = tid; u < kNRel * kDH * kDH; u += 256) {
    const int e = u >> 10, pqi = u & 1023;
    const int p = pqi >> 5, cc = pqi & 31;
    WatL[e][p * WS + cc] = rel_att[(size_t)(e * kHeads + mh) * (kDH * kDH) + pqi];
    WmT [e][cc * WS + p] = rel_msg[(size_t)(e * kHeads + mh) * (kDH * kDH) + pqi];
  }
  // k/v tiles: 16 pixels x 5 nodes x 32 channels for this head.
  for (int u = tid; u < 16 * kL * kDH; u += 256) {
    const int pix = u / (kL * kDH);
    const int rem = u - pix * (kL * kDH);
    const int i = rem >> 5, c = rem & 31;
    const size_t g = ((size_t)(bb * kHW + pix0 + pix) * kL + i) * kInner + mh * kDH + c;
    const int s = pix * TS + i * kDH + c;
    KT[s] = k[g];
    VT[s] = v[g];
  }
  __syncthreads();

  // ---- Phase 1: attention logits (wave-uniform pair loop; EXEC all-ones) ----
  for (int pr = wave; pr < kL * kL; pr += 8) {
    const int i = pr / kL, j = pr % kL;
    const int e = typesS[i] * kNT + typesS[j];
    v8f c0 = zero8(), c1 = zero8();
    #pragma unroll
    for (int kk = 0; kk < kDH; kk += 4) {
      const v2f a  = *(const v2f*)(&KT[mm * TS + j * kDH + kk + kb2]);
      const v2f b0 = *(const v2f*)(&WatL[e][mm * WS + kk + kb2]);        // B[q][p]=Wat[p][q]
      const v2f b1 = *(const v2f*)(&WatL[e][(16 + mm) * WS + kk + kb2]);
      c0 = WMMA_F32(a, b0, c0);
      c1 = WMMA_F32(a, b1, c1);
    }
    // att[pix,i,j] = scale * sum_c q[pix,i,c] * t[pix,c]; reduce across 16 lanes.
    #pragma unroll
    for (int r = 0; r < 8; ++r) {
      const int row = r + 8 * hi;
      const size_t gq = ((size_t)(bb * kHW + pix0 + row) * kL + i) * kInner + mh * kDH;
      float s = q[gq + mm] * c0[r] + q[gq + 16 + mm] * c1[r];
      s += __shfl_xor(s, 1, 32);
      s += __shfl_xor(s, 2, 32);
      s += __shfl_xor(s, 4, 32);
      s += __shfl_xor(s, 8, 32);
      if (mm == 0) attS[row][i][j] = s * 0.17677669529663688f;  // 32^-0.5
    }
  }
  __syncthreads();

  // ---- Phase 2: mask + softmax over j (one thread per (pixel,i) row) ----
  if (tid < 16 * kL) {
    const int pix = tid / kL, i = tid % kL;
    const int gp = pix0 + pix, hh = gp >> 6, ww = gp & 63;
    const float* mrow = mask + ((((size_t)bb * 64 + hh) * 64 + ww) * kL + i) * kL;
    float vals[kL];
    float mx = -__builtin_inff();
    #pragma unroll
    for (int j = 0; j < kL; ++j) {
      float a = attS[pix][i][j];
      if (mrow[j] == 0.0f) a = -__builtin_inff();
      vals[j] = a;
      mx = fmaxf(mx, a);
    }
    float se = 0.0f;
    #pragma unroll
    for (int j = 0; j < kL; ++j) { const float ev = __expf(vals[j] - mx); vals[j] = ev; se += ev; }
    const float inv = 1.0f / se;
    #pragma unroll
    for (int j = 0; j < kL; ++j) attS[pix][i][j] = vals[j] * inv;
  }
  __syncthreads();

  // ---- Phase 3: message aggregation (wave i handles output node i) ----
  if (wave < kL) {
    const int i = wave;
    v8f o0 = zero8(), o1 = zero8();
    for (int j = 0; j < kL; ++j) {
      const int e = typesS[i] * kNT + typesS[j];
      v8f c0 = zero8(), c1 = zero8();
      #pragma unroll
      for (int kk = 0; kk < kDH; kk += 4) {
        const v2f a  = *(const v2f*)(&VT[mm * TS + j * kDH + kk + kb2]);
        const v2f b0 = *(const v2f*)(&WmT[e][mm * WS + kk + kb2]);        // B[p][c]=Wmsg[p][c]
        const v2f b1 = *(const v2f*)(&WmT[e][(16 + mm) * WS + kk + kb2]);
        c0 = WMMA_F32(a, b0, c0);
        c1 = WMMA_F32(a, b1, c1);
      }
      #pragma unroll
      for (int r = 0; r < 8; ++r) {
        const float aw = attS[r + 8 * hi][i][j];   // broadcast read per half-wave
        o0[r] += aw * c0[r];
        o1[r] += aw * c1[r];
      }
    }
    #pragma unroll
    for (int r = 0; r < 8; ++r) {
      const int row = r + 8 * hi;
      const size_t g = ((size_t)(bb * kHW + pix0 + row) * kL + i) * kInner + mh * kDH;
      hm[g + mm]      = o0[r];
      hm[g + 16 + mm] = o1[r];
    }
  }
}

// ---------------------------------------------------------------------------
// Kernel 3: output projection. A rows strided (hm is [B][HW][L][INNER]);
// output contiguous in reference order [B][L][H][W][DIM].
// ---------------------------------------------------------------------------
__global__ __launch_bounds__(256) void out_proj_kernel(
    const float* __restrict__ hm, const float* __restrict__ pe,
    const float* __restrict__ Wa, const float* __restrict__ ba,
    float* __restrict__ out)
{
  constexpr int AS = 36;
  constexpr int BS = 34;
  __shared__ __align__(16) float Alds[64 * AS];
  __shared__ __align__(16) float Blds[256 * BS];

  const int bl   = blockIdx.y;
  const int bb   = bl / kL, ll = bl % kL;
  const int row0 = blockIdx.x * 64;
  const int typ  = (int)pe[(bb * kL + ll) * 3 + 2];

  const float* W    = Wa + (size_t)typ * kInner * kDim;
  const float* bias = ba + typ * kDim;

  const int tid  = threadIdx.x;
  const int wave = tid >> 5, lane = tid & 31;
  const int rt     = wave & 3;
  const int ntBase = (wave >> 2) * 8;
  const int mm = lane & 15, hi = lane >> 4, kb2 = hi * 2;

  const float* Abase = hm + ((size_t)(bb * kHW + row0) * kL + ll) * kInner;
  constexpr unsigned int Astride = kL * kInner;   // 1280 floats between pixel rows
  const unsigned int aLds = lds_offset(&Alds[0]);

  v8f acc[8];
  #pragma unroll
  for (int t = 0; t < 8; ++t) acc[t] = zero8();

  for (int kb = 0; kb < kInner; kb += 32) {
    __syncthreads();
    if (wave == 0) tdm_load_a_tile(aLds, Abase + kb, Astride, 64);
    for (int u = tid; u < 16 * 64; u += 256) {
      const int kp = u >> 6, c4 = (u & 63) << 2, k2 = kp * 2;
      const float4 r0 = *(const float4*)(W + (size_t)(kb + k2) * kDim + c4);
      const float4 r1 = *(const float4*)(W + (size_t)(kb + k2 + 1) * kDim + c4);
      v2f p0 = {r0.x, r1.x}, p1 = {r0.y, r1.y}, p2 = {r0.z, r1.z}, p3 = {r0.w, r1.w};
      *(v2f*)(&Blds[(c4 + 0) * BS + k2]) = p0;
      *(v2f*)(&Blds[(c4 + 1) * BS + k2]) = p1;
      *(v2f*)(&Blds[(c4 + 2) * BS + k2]) = p2;
      *(v2f*)(&Blds[(c4 + 3) * BS + k2]) = p3;
    }
    if (wave == 0) __builtin_amdgcn_s_wait_tensorcnt(0);
    __syncthreads();

    #pragma unroll
    for (int kk = 0; kk < 32; kk += 4) {
      const v2f a = *(const v2f*)(&Alds[(rt * 16 + mm) * AS + kk + kb2]);
      #pragma unroll
      for (int t = 0; t < 8; ++t) {
        const int ncol = (ntBase + t) * 16 + mm;
        const v2f bf = *(const v2f*)(&Blds[ncol * BS + kk + kb2]);
        acc[t] = WMMA_F32(a, bf, acc[t]);
      }
    }
  }

  #pragma unroll
  for (int t = 0; t < 8; ++t) {
    const int col = (ntBase + t) * 16 + mm;
    const float bia = bias[col];
    #pragma unroll
    for (int r = 0; r < 8; ++r) {
      const int row = row0 + rt * 16 + r + 8 * hi;
      out[((size_t)(bb * kL + ll) * kHW + row) * kDim + col] = acc[t][r] + bia;
    }
  }
}

// ---------------------------------------------------------------------------
extern "C" void kernel_launch(void* const* d_in, const int* in_sizes, int n_in,
                              void* d_out, int out_size, void* d_ws, size_t ws_size,
                              hipStream_t stream) {
  (void)in_sizes; (void)n_in; (void)out_size;

  const float* x    = (const float*)d_in[0];
  const float* mask = (const float*)d_in[1];
  const float* pe   = (const float*)d_in[2];
  const float* Wq   = (const float*)d_in[3];
  const float* bq   = (const float*)d_in[4];
  const float* Wk   = (const float*)d_in[5];
  const float* bk   = (const float*)d_in[6];
  const float* Wv   = (const float*)d_in[7];
  const float* bv   = (const float*)d_in[8];
  const float* Wa   = (const float*)d_in[9];
  const float* ba   = (const float*)d_in[10];
  const float* ra   = (const float*)d_in[11];
  const float* rm   = (const float*)d_in[12];
  float* out = (float*)d_out;

  const size_t tensorElems = (size_t)kB * kHW * kL * kInner;  // 10,485,760 floats
  if (ws_size < 3 * tensorElems * sizeof(float)) return;      // need 126 MB scratch

  float* qws = (float*)d_ws;
  float* kws = qws + tensorElems;
  float* vws = kws + tensorElems;
  float* hm  = qws;  // merged-heads buffer aliases q (see attn_kernel note)

  qkv_proj_kernel<<<dim3(kHW / 64, kB * kL, 3), 256, 0, stream>>>(
      x, pe, Wq, bq, qws, Wk, bk, kws, Wv, bv, vws);

  attn_kernel<<<dim3(kHW / 16, kHeads, kB), 256, 0, stream>>>(
      qws, kws, vws, mask, pe, ra, rm, hm);

  out_proj_kernel<<<dim3(kHW / 64, kB * kL, 1), 256, 0, stream>>>(
      hm, pe, Wa, ba, out);
}